// GridEncoder_68959994904855
// MI455X (gfx1250) — compile-verified
//
#include <hip/hip_runtime.h>
#include <cmath>
#include <cstdint>
#include <cstddef>

// ---------------------------------------------------------------------------
// GridEncoder (Instant-NGP hash grid, forward) for MI455X / gfx1250.
//
// B = 1M points, 16 levels, 2 features/level. Gather-bound: ~128M random
// 8-byte gathers from a 49MB table (L2-resident on MI455X's 192MB L2).
// One thread per (point, level); 8 float2 gathers per thread; nontemporal
// output stores so the streamed 128MB output does not evict the table.
// Inputs staged into LDS via CDNA5 async-to-LDS DMA (ASYNCcnt-tracked).
// ---------------------------------------------------------------------------

#define NLV 16   // levels
#define PPB 16   // points per block
#define TPB 256  // threads per block  (PPB * NLV)

typedef float v2f __attribute__((ext_vector_type(2)));

struct alignas(16) LevelParam {
  float    scale;   // exp2(l*S)*16 - 1  (float32, matches reference)
  unsigned G;       // grid verts per dim
  unsigned G2;      // G*G
  unsigned mask;    // hashmap-1 (hashmap is 2^19 for hashed levels)
  unsigned offset;  // level offset into embedding table
  unsigned hashed;  // 1 if hashed level, 0 if dense
  unsigned pad0, pad1;                 // pad to 32B -> b128 LDS traffic
};
struct Params { LevelParam lp[NLV]; };

#if __has_builtin(__builtin_amdgcn_global_load_async_to_lds_b32) && \
    __has_builtin(__builtin_amdgcn_s_wait_asynccnt)
#define USE_ASYNC_LDS 1
typedef __attribute__((address_space(1))) int* as1_i32p;
typedef __attribute__((address_space(3))) int* as3_i32p;
#endif

__global__ __launch_bounds__(TPB)
void grid_encode_kernel(const float* __restrict__ inp,
                        const float* __restrict__ emb,
                        float* __restrict__ out,
                        Params P, int npts)
{
  __shared__ float      xs[PPB * 3];
  __shared__ LevelParam slp[NLV];

  const int tid   = threadIdx.x;
  const int level = tid & (NLV - 1);
  const int psub  = tid >> 4;
  const int p     = blockIdx.x * PPB + psub;

  // --- Stage this block's 48 input floats into LDS (async DMA on CDNA5) ---
  const float* src = inp + (size_t)blockIdx.x * (PPB * 3);
#ifdef USE_ASYNC_LDS
  if (tid < PPB * 3) {
    __builtin_amdgcn_global_load_async_to_lds_b32(
        (as1_i32p)(uintptr_t)(src + tid),
        (as3_i32p)(&xs[tid]), 0, 0);
  }
#else
  if (tid < PPB * 3) xs[tid] = src[tid];
#endif

  // --- Stage per-level params into LDS from a single lane with constant
  //     indices only: one branch, pure s_load + ds_store (no select tree,
  //     no scratch copy of the by-value struct). ---
  if (tid == 0) {
#pragma unroll
    for (int l = 0; l < NLV; ++l) slp[l] = P.lp[l];
  }

#ifdef USE_ASYNC_LDS
  __builtin_amdgcn_s_wait_asynccnt(0);
#endif
  __syncthreads();

  if (p >= npts) return;

  const LevelParam L = slp[level];

  const float x0 = xs[psub * 3 + 0];
  const float x1 = xs[psub * 3 + 1];
  const float x2 = xs[psub * 3 + 2];

  // Position / interpolation weights — keep mul+add separate (no FMA
  // contraction) to match the reference's computation order.
  float f0, f1, f2;
  unsigned u0, u1, u2;
  {
#pragma clang fp contract(off)
    const float y0 = (x0 + 1.0f) * 0.5f;
    const float y1 = (x1 + 1.0f) * 0.5f;
    const float y2 = (x2 + 1.0f) * 0.5f;
    const float p0 = y0 * L.scale + 0.5f;
    const float p1 = y1 * L.scale + 0.5f;
    const float p2 = y2 * L.scale + 0.5f;
    const float g0 = floorf(p0), g1 = floorf(p1), g2 = floorf(p2);
    f0 = p0 - g0; f1 = p1 - g1; f2 = p2 - g2;
    u0 = (unsigned)(int)g0; u1 = (unsigned)(int)g1; u2 = (unsigned)(int)g2;
  }

  // Per-dimension index terms (2 values per dim -> 8 corners by combination).
  unsigned t0[2], t1[2], t2[2];
  if (L.hashed) {
    t0[0] = u0;                       t0[1] = u0 + 1u;
    t1[0] = u1 * 2654435761u;         t1[1] = (u1 + 1u) * 2654435761u;
    t2[0] = u2 * 805459861u;          t2[1] = (u2 + 1u) * 805459861u;
  } else {
    t0[0] = u0;                       t0[1] = u0 + 1u;
    t1[0] = u1 * L.G;                 t1[1] = t1[0] + L.G;
    t2[0] = u2 * L.G2;                t2[1] = t2[0] + L.G2;
  }

  // Compute all 8 indices first so the 8 gathers can be issued back-to-back.
  unsigned idx[8];
#pragma unroll
  for (int c = 0; c < 8; ++c) {
    const int d0 = (c >> 2) & 1, d1 = (c >> 1) & 1, d2 = c & 1;
    // Hashed: xor-of-primes then power-of-two mask (hashmap == 2^19).
    // Dense: row-major index, provably < hashmap -> no modulo needed.
    const unsigned v = L.hashed ? ((t0[d0] ^ t1[d1] ^ t2[d2]) & L.mask)
                                : (t0[d0] + t1[d1] + t2[d2]);
    idx[c] = v + L.offset;
  }

  const v2f* __restrict__ E = (const v2f*)emb;
  v2f e[8];
#pragma unroll
  for (int c = 0; c < 8; ++c) e[c] = E[(size_t)idx[c]];

  const float w0[2] = {1.0f - f0, f0};
  const float w1[2] = {1.0f - f1, f1};
  const float w2[2] = {1.0f - f2, f2};
  float a0 = 0.0f, a1 = 0.0f;
#pragma unroll
  for (int c = 0; c < 8; ++c) {
    const int d0 = (c >> 2) & 1, d1 = (c >> 1) & 1, d2 = c & 1;
    const float w = w0[d0] * w1[d1] * w2[d2];
    a0 += w * e[c].x;
    a1 += w * e[c].y;
  }

  // Output row-major [B, 32]; 16 lanes of one point cover a contiguous 128B
  // row. Nontemporal: keep L2 for the embedding table, stream the output.
  v2f o; o.x = a0; o.y = a1;
  __builtin_nontemporal_store(o, (v2f*)out + (size_t)p * NLV + level);
}

extern "C" void kernel_launch(void* const* d_in, const int* in_sizes, int n_in,
                              void* d_out, int out_size, void* d_ws, size_t ws_size,
                              hipStream_t stream) {
  const float* inp = (const float*)d_in[0];
  const float* emb = (const float*)d_in[1];
  float* out = (float*)d_out;
  const int npts = in_sizes[0] / 3;
  if (npts <= 0) return;

  // Replicate the reference's level-offset computation exactly (double math,
  // numpy-style ceil), so the kernel sees precomputed per-level constants.
  Params P;
  const double PLS = exp2(log2(2048.0 / 16.0) / 15.0);
  const double S   = log2(PLS);
  unsigned off = 0;
  for (int l = 0; l < NLV; ++l) {
    const double xr = 16.0 * pow(PLS, (double)l);
    const unsigned long long res = (unsigned long long)ceil(xr) + 1ull;
    unsigned long long n = res * res * res;
    if (n > (1ull << 19)) n = (1ull << 19);
    n = ((n + 7ull) / 8ull) * 8ull;                       // ceil to mult of 8
    const float scale = (float)(exp2((double)l * S) * 16.0 - 1.0);
    const unsigned G = (unsigned)ceilf(scale) + 2u;
    const unsigned long long G3 = (unsigned long long)G * G * G;
    P.lp[l].scale  = scale;
    P.lp[l].G      = G;
    P.lp[l].G2     = G * G;
    P.lp[l].mask   = (unsigned)(n - 1ull);                // pow2 when hashed
    P.lp[l].offset = off;
    P.lp[l].hashed = (G3 > n) ? 1u : 0u;
    P.lp[l].pad0   = 0u;
    P.lp[l].pad1   = 0u;
    off += (unsigned)n;
  }

  const int blocks = (npts + PPB - 1) / PPB;
  grid_encode_kernel<<<blocks, TPB, 0, stream>>>(inp, emb, out, P, npts);
}